// DifferentiableMinVarianceLayer_55396488184583
// MI455X (gfx1250) — compile-verified
//
#include <hip/hip_runtime.h>

// CDNA5 / gfx1250: wave32, WMMA f32 16x16x4, 320KB LDS per WGP.
typedef float v2f __attribute__((ext_vector_type(2)));
typedef float v8f __attribute__((ext_vector_type(8)));

#define NN       100            // real problem size
#define NP       112            // padded to 7 x 16 WMMA tiles
#define ST       114            // LDS row stride (floats): bank-friendly, 8B-aligned rows
#define WPB      6              // waves (=batches) per block; 6*51.6KB LDS <= 320KB
#define TPB      (32 * WPB)
#define EPSJ     1e-6f
#define N_ITERS  500
#define PROJ_IT  24
#define LDS_PER_WAVE (NP * ST + 128)   // matrix + w-vector region (floats)

__device__ __forceinline__ float wave_sum(float x) {
#pragma unroll
  for (int m = 16; m > 0; m >>= 1) x += __shfl_xor(x, m, 32);
  return x;
}

__global__ void __launch_bounds__(TPB)
minvar_pgd_kernel(const float* __restrict__ sigma, float* __restrict__ out, int nbatch) {
  extern __shared__ float lds[];
  const int lane = threadIdx.x & 31;
  const int wave = threadIdx.x >> 5;
  const int b = blockIdx.x * WPB + wave;
  if (b >= nbatch) return;                    // wave-uniform; no block barriers used

  float* S = lds + wave * LDS_PER_WAVE;       // 112 x 114 matrix (L, then M)
  float* W = S + NP * ST;                     // w vector (128 floats)
  const float* sig = sigma + (size_t)b * NN * NN;

  // ---------- Phase 1: load + symmetrize + eps*I, pad to 112 with unit diag ----------
  {
    const int j0 = lane * 4;                  // lanes 0..27 cover 112 columns
    for (int i = 0; i < NP; ++i) {
      if (j0 < NP) {
#pragma unroll
        for (int q = 0; q < 4; ++q) {
          const int j = j0 + q;
          float v;
          if (i < NN && j < NN) {
            v = 0.5f * (sig[i * NN + j] + sig[j * NN + i]);
            if (i == j) v += EPSJ;
          } else {
            v = (i == j) ? 1.0f : 0.0f;       // identity pad -> Cholesky stays clean
          }
          S[i * ST + j] = v;
        }
      }
    }
  }

  // ---------- Phase 2: in-place Cholesky (lower), wave-synchronous ----------
  for (int k = 0; k < NP; ++k) {
    const float dkk = S[k * ST + k];          // broadcast LDS read
    const float inv = rsqrtf(dkk);
    if (lane == 0) S[k * ST + k] = dkk * inv; // sqrt(dkk)
    for (int i = k + 1 + lane; i < NP; i += 32) S[i * ST + k] *= inv;
    for (int j = k + 1; j < NP; ++j) {
      const float ljk = S[j * ST + k];
      for (int i = j + lane; i < NP; i += 32)
        S[i * ST + j] -= S[i * ST + k] * ljk; // trailing update, lower only
    }
  }

  // ---------- Phase 3: M = L^T L via V_WMMA_F32_16X16X4_F32, into upper triangle ----------
  // Fragment loads are guarded (col <= k) so only valid lower-tri L is read;
  // strictly-upper M tiles can therefore be written immediately. Diagonal tiles
  // are stashed in VGPRs and written after all L reads complete.
  const int m16 = lane & 15;
  const int hi8 = (lane < 16) ? 0 : 8;
  const int kap = (lane < 16) ? 0 : 2;        // A/B f32 frag: lanes 0-15 K=0,1; 16-31 K=2,3
  float dstash[7][8];
#pragma unroll
  for (int ti = 0; ti < 7; ++ti) {
#pragma unroll
    for (int tj = ti; tj < 7; ++tj) {
      v8f acc = {0.f, 0.f, 0.f, 0.f, 0.f, 0.f, 0.f, 0.f};
      const int ca = ti * 16 + m16;           // A column of L (row of L^T)
      const int cb = tj * 16 + m16;           // B column of L
      for (int k0 = tj * 16; k0 < NP; k0 += 4) {   // k-blocks below tj*16 are all-zero
        const int ka = k0 + kap;
        const float a0 = S[ka * ST + ca], a1 = S[(ka + 1) * ST + ca];
        const float b0 = S[ka * ST + cb], b1 = S[(ka + 1) * ST + cb];
        v2f A, Bm;
        A.x  = (ca <= ka    ) ? a0 : 0.0f;    // L is lower: L[k][c] valid iff c <= k
        A.y  = (ca <= ka + 1) ? a1 : 0.0f;
        Bm.x = (cb <= ka    ) ? b0 : 0.0f;
        Bm.y = (cb <= ka + 1) ? b1 : 0.0f;
        acc = __builtin_amdgcn_wmma_f32_16x16x4_f32(
            false, A, false, Bm, (short)0, acc, false, false);
      }
      if (ti == tj) {
#pragma unroll
        for (int r = 0; r < 8; ++r) dstash[ti][r] = acc[r];
      } else {
#pragma unroll
        for (int r = 0; r < 8; ++r)
          S[(ti * 16 + hi8 + r) * ST + tj * 16 + m16] = acc[r];  // strictly upper
      }
    }
  }
#pragma unroll
  for (int t = 0; t < 7; ++t) {               // diagonal tiles: write j >= i part
#pragma unroll
    for (int r = 0; r < 8; ++r) {
      const int i = t * 16 + hi8 + r;
      const int j = t * 16 + m16;
      if (j >= i) S[i * ST + j] = dstash[t][r];
    }
  }

  // ---------- Phase 4: mirror upper -> lower (L is dead; M becomes full dense) ----------
  for (int i = 0; i < NP - 1; ++i)
    for (int j = i + 1 + lane; j < NP; j += 32)
      S[j * ST + i] = S[i * ST + j];

  // ---------- Phase 5: trace (real 100 diag only) and step size ----------
  float tr = 0.0f;
  for (int i = lane; i < NN; i += 32) tr += S[i * ST + i];
  tr = wave_sum(tr);
  const float step2 = 1.0f / fmaxf(tr, EPSJ);  // = 2 * (0.5 / max(tr,eps))

  // ---------- Phase 6: init w ----------
  const int i0 = lane, i1 = lane + 32, i2 = lane + 64;
  const int i3 = (lane + 96 < NP) ? (lane + 96) : 0;  // clamp OOB rows (lanes >= 16)
  const bool v3ok = (lane < 4);               // row lane+96 < 100 only for lanes 0..3
  float w0 = 1.0f / NN, w1 = w0, w2 = w0, w3 = v3ok ? w0 : 0.0f;
  for (int j = lane; j < 128; j += 32) W[j] = 0.0f;
  W[i0] = w0; W[i1] = w1; W[i2] = w2;
  if (v3ok) W[lane + 96] = w3;

  const float* R0 = S + i0 * ST;
  const float* R1 = S + i1 * ST;
  const float* R2 = S + i2 * ST;
  const float* R3 = S + i3 * ST;

  // ---------- Phase 7: 500 projected-gradient iterations ----------
#pragma unroll 1
  for (int it = 0; it < N_ITERS; ++it) {
    // y = M w : 4 rows per lane, float2 LDS loads, conflict-free (stride 114)
    float y0 = 0.f, y1 = 0.f, y2 = 0.f, y3 = 0.f;
#pragma unroll 4
    for (int j = 0; j < NP; j += 2) {
      const float2 wv = *(const float2*)(W + j);   // broadcast
      const float2 m0 = *(const float2*)(R0 + j);
      const float2 m1 = *(const float2*)(R1 + j);
      const float2 m2 = *(const float2*)(R2 + j);
      const float2 m3 = *(const float2*)(R3 + j);
      y0 += m0.x * wv.x + m0.y * wv.y;
      y1 += m1.x * wv.x + m1.y * wv.y;
      y2 += m2.x * wv.x + m2.y * wv.y;
      y3 += m3.x * wv.x + m3.y * wv.y;
    }
    const float v0 = w0 - step2 * y0;
    const float v1 = w1 - step2 * y1;
    const float v2 = w2 - step2 * y2;
    const float v3 = v3ok ? (w3 - step2 * y3) : -1e30f;  // sentinel: never active

    // Simplex projection: Newton/Michelot fixed point on tau (exact, sort-free)
    float tau = (wave_sum(v0 + v1 + v2 + (v3ok ? v3 : 0.0f)) - 1.0f) * (1.0f / NN);
#pragma unroll 1
    for (int t = 0; t < PROJ_IT; ++t) {
      const float a0 = (v0 > tau) ? 1.0f : 0.0f;
      const float a1 = (v1 > tau) ? 1.0f : 0.0f;
      const float a2 = (v2 > tau) ? 1.0f : 0.0f;
      const float a3 = (v3 > tau) ? 1.0f : 0.0f;
      float ls = a0 * v0 + a1 * v1 + a2 * v2 + a3 * v3;
      float lc = a0 + a1 + a2 + a3;
      ls = wave_sum(ls);
      lc = wave_sum(lc);
      tau = (ls - 1.0f) / fmaxf(lc, 1.0f);
    }
    w0 = fmaxf(v0 - tau, 0.0f);
    w1 = fmaxf(v1 - tau, 0.0f);
    w2 = fmaxf(v2 - tau, 0.0f);
    w3 = v3ok ? fmaxf(v3 - tau, 0.0f) : 0.0f;
    W[i0] = w0; W[i1] = w1; W[i2] = w2;
    if (v3ok) W[lane + 96] = w3;               // pad entries of W stay 0
  }

  // ---------- Phase 8: write output [B, 100, 1] ----------
  float* o = out + (size_t)b * NN;
  o[i0] = w0; o[i1] = w1; o[i2] = w2;
  if (v3ok) o[lane + 96] = w3;
}

extern "C" void kernel_launch(void* const* d_in, const int* in_sizes, int n_in,
                              void* d_out, int out_size, void* d_ws, size_t ws_size,
                              hipStream_t stream) {
  (void)n_in; (void)d_ws; (void)ws_size; (void)out_size;
  const float* sigma = (const float*)d_in[0];
  float* out = (float*)d_out;
  const int nbatch = in_sizes[0] / (NN * NN);          // 4096
  const int blocks = (nbatch + WPB - 1) / WPB;
  const size_t shmem = (size_t)WPB * LDS_PER_WAVE * sizeof(float);  // ~309.5 KB
  // Opt in to large dynamic LDS (deterministic host-side call; safe under graph capture).
  (void)hipFuncSetAttribute((const void*)minvar_pgd_kernel,
                            hipFuncAttributeMaxDynamicSharedMemorySize, (int)shmem);
  minvar_pgd_kernel<<<blocks, TPB, shmem, stream>>>(sigma, out, nbatch);
}